// GLayer_39633958208176
// MI455X (gfx1250) — compile-verified
//
#include <hip/hip_runtime.h>
#include <hip/hip_bf16.h>

typedef __attribute__((ext_vector_type(2))) float v2f;
typedef __attribute__((ext_vector_type(8))) float v8f;

#define G_GAMMA 0.01f
#define G_ALPHA 1.0f
#define G_TOL 1e-6f
#define G_NUM_ITER 100
#define G_GUESS_RATIO (1.0f - 0.001f)
#define DD 512
#define KK 32
#define PP 528 /* K*(K+1)/2 */

// ---------------------------------------------------------------------------
// Kernel 1: out = Z_all * Z_NORM + Z_MU  (Z_NORM=1, Z_MU=0 -> copy).
// Also provides the Z term read back by main_kernel stage 7.
// ---------------------------------------------------------------------------
__global__ void copy_kernel(const float* __restrict__ z, float* __restrict__ out,
                            long n4) {
  long i = (long)blockIdx.x * blockDim.x + threadIdx.x;
  long stride = (long)gridDim.x * blockDim.x;
  const float4* zi = (const float4*)z;
  float4* oo = (float4*)out;
  for (; i < n4; i += stride) oo[i] = zi[i];
}

// ---------------------------------------------------------------------------
// Kernel 2 (1 block, 512 threads): x_muU, x_cU, change -> workspace.
//   ws[0..31]   = x_muU
//   ws[32..63]  = x_cU
//   ws[64..575] = change = x_c - z_mu - (x_cU - x_muU) @ U.T
// ---------------------------------------------------------------------------
__global__ void prep_kernel(const float* __restrict__ U,
                            const float* __restrict__ zmu,
                            const float* __restrict__ xc,
                            float* __restrict__ ws) {
  __shared__ float sxm[KK], sxc[KK];
  int t = threadIdx.x;
  if (t < KK) {
    float s = 0.f;
    for (int d = 0; d < DD; ++d) s += zmu[d] * U[d * KK + t];
    sxm[t] = s;
    ws[t] = s;
  } else if (t < 2 * KK) {
    int k = t - KK;
    float s = 0.f;
    for (int d = 0; d < DD; ++d) s += xc[d] * U[d * KK + k];
    sxc[k] = s;
    ws[KK + k] = s;
  }
  __syncthreads();
  if (t < DD) {
    float s = xc[t] - zmu[t];
    for (int k = 0; k < KK; ++k) s -= (sxc[k] - sxm[k]) * U[t * KK + k];
    ws[2 * KK + t] = s;
  }
}

// ---------------------------------------------------------------------------
// Kernel 3: 16 rows per block, 256 threads (8 wave32).
// LDS overlay: Zs (16x516, live through ZU stage) shares storage with
// Hs (16x532, live from H build through the main GEMM).
// ---------------------------------------------------------------------------
__global__ void __launch_bounds__(256) main_kernel(
    const float* __restrict__ Zall, const float* __restrict__ U,
    const float* __restrict__ V, const float* __restrict__ zmu_g,
    const int* __restrict__ cidx_g, const float* __restrict__ ws,
    float* __restrict__ out, int N) {
  __shared__ __align__(16) char smem_u[16 * 532 * 4];  // max(Zs, Hs) = 34048 B
  float (*Zs)[516] = reinterpret_cast<float(*)[516]>(smem_u);
  float (*Hs)[532] = reinterpret_cast<float(*)[532]>(smem_u);

  __shared__ float zu[16][33];    // ZU, then reused as coord (padded)
  __shared__ float chg[DD];
  __shared__ float zmu[DD];
  __shared__ float xmu[KK], xcu[KK];
  __shared__ float zn2[16], zun2[16], kern[16];
  __shared__ int lutI[PP], lutJ[PP];
  __shared__ int cidx[16];
  __shared__ int validm[16];

  const int tid = threadIdx.x;
  const int wv = tid >> 5;
  const int lane = tid & 31;
  const int l16 = lane & 15;
  const int hi = lane >> 4;       // 0: K pair {0,1}, 1: K pair {2,3} of a step
  const int row0 = blockIdx.x * 16;

  // ---- stage 0: constants, LUT, row indices -------------------------------
  for (int i = tid; i < DD; i += 256) {
    chg[i] = ws[2 * KK + i];
    zmu[i] = zmu_g[i];
  }
  if (tid < KK) {
    xmu[tid] = ws[tid];
    xcu[tid] = ws[KK + tid];
  }
  if (tid < 16) {
    int r = row0 + tid;
    validm[tid] = (r < N);
    cidx[tid] = (r < N) ? cidx_g[r] : cidx_g[0];
    zn2[tid] = 0.f;
  }
  for (int j = tid; j < PP; j += 256) {  // triu_indices(32) row-major
    int ii = 0, base = 0;
    while (base + (KK - ii) <= j) { base += (KK - ii); ++ii; }
    lutI[j] = ii;
    lutJ[j] = ii + (j - base);
  }
  __syncthreads();

  // ---- stage 1: load Z tile, accumulate Z_norm2 ---------------------------
  for (int s = tid; s < 16 * (DD / 4); s += 256) {
    int r = s >> 7;
    int c4 = s & 127;
    float4 v = ((const float4*)Zall)[(size_t)cidx[r] * (DD / 4) + c4];
    *((float4*)&Zs[r][c4 * 4]) = v;
    int c = c4 * 4;
    float d0 = v.x - zmu[c + 0], d1 = v.y - zmu[c + 1];
    float d2 = v.z - zmu[c + 2], d3 = v.w - zmu[c + 3];
    atomicAdd(&zn2[r], d0 * d0 + d1 * d1 + d2 * d2 + d3 * d3);
  }
  __syncthreads();

  // ---- stage 2: ZU = Ztile(16x512) @ U(512x32) via f32 WMMA (waves 0,1) ---
  if (wv < 2) {
    int n0 = wv * 16;
    v8f acc = {};
    for (int kk = 0; kk < DD; kk += 4) {
      int k0 = kk + 2 * hi;
      v2f a, b;
      a.x = Zs[l16][k0];
      a.y = Zs[l16][k0 + 1];
      b.x = U[(size_t)k0 * KK + n0 + l16];
      b.y = U[(size_t)(k0 + 1) * KK + n0 + l16];
      acc = __builtin_amdgcn_wmma_f32_16x16x4_f32(false, a, false, b,
                                                  (short)0, acc, false, false);
    }
    int mb = hi * 8;
    for (int r = 0; r < 8; ++r) zu[mb + r][n0 + l16] = acc[r];
  }
  __syncthreads();

  // ---- stage 3: ZU_norm2 --------------------------------------------------
  if (tid < 16) {
    float s = 0.f;
    for (int k = 0; k < KK; ++k) {
      float d = zu[tid][k] - xmu[k];
      s += d * d;
    }
    zun2[tid] = s;
  }
  __syncthreads();

  // ---- stage 4: coord = ZU - x_cU (in place) ------------------------------
  for (int s = tid; s < 16 * KK; s += 256) {
    int m = s >> 5, k = s & 31;
    zu[m][k] = zu[m][k] - xcu[k];
  }
  __syncthreads();

  // ---- stage 5a: secant root solve + kernel value (16 lanes) --------------
  if (tid < 16) {
    float zun = zun2[tid];
    float zup = zn2[tid] - zun;           // ZUperp_norm2
    float eg = expf(-G_GAMMA * zun);
    float xm2 = zup * G_GUESS_RATIO, xm1 = zup;
    for (int it = 0; it < G_NUM_ITER; ++it) {
      float t1 = 1.f - G_ALPHA * eg * expf(-G_GAMMA * xm1);
      float f1 = t1 * t1 * xm1 - zup;
      float t2 = 1.f - G_ALPHA * eg * expf(-G_GAMMA * xm2);
      float f2 = t2 * t2 * xm2 - zup;
      float fd = f1 - f2;
      if (fabsf(fd) < G_TOL) fd = (fd >= 0.f) ? G_TOL : -G_TOL;
      float x = xm1 - f1 * (xm1 - xm2) / fd;
      xm2 = xm1;
      xm1 = x;
    }
    kern[tid] = G_ALPHA * expf(-G_GAMMA * (zun + xm1));
  }

  // ---- stage 5b: build H tile ONCE (overlays Zs; Zs dead after stage 2) ---
  // Note: no barrier needed between 5a and 5b w.r.t. each other (disjoint
  // data), but H overwrites Zs which stage 2 read -> already separated by the
  // barrier after stage 4. H reads zu (coord) -> separated by same barrier.
  for (int s = tid; s < 16 * PP; s += 256) {
    int m = s / PP;
    int j = s - m * PP;
    Hs[m][j] = zu[m][lutI[j]] * zu[m][lutJ[j]];
  }
  __syncthreads();

  // ---- stage 6: Out(16x512) = H(16x528) @ V^T; 4 col-tiles per wave -------
  const int n0b = wv * 64;
  v8f acc[4] = {};
  for (int kk = 0; kk < PP; kk += 4) {
    int k0 = kk + 2 * hi;
    v2f a = *(const v2f*)&Hs[l16][k0];    // single 8B LDS read per step
    const float* vp = V + (size_t)(n0b + l16) * PP + k0;  // B[k][n] = V[n][k]
#pragma unroll
    for (int t = 0; t < 4; ++t) {
      v2f b;
      b.x = vp[(size_t)t * 16 * PP];
      b.y = vp[(size_t)t * 16 * PP + 1];
      acc[t] = __builtin_amdgcn_wmma_f32_16x16x4_f32(false, a, false, b,
                                                     (short)0, acc[t], false, false);
    }
  }

  // ---- stage 7: Xhat = Z + kernel*(change + acc); Z read back from out ----
  const int mb = hi * 8;
#pragma unroll
  for (int t = 0; t < 4; ++t) {
    int col = n0b + t * 16 + l16;
#pragma unroll
    for (int r = 0; r < 8; ++r) {
      int m = mb + r;
      if (validm[m]) {
        size_t o = (size_t)cidx[m] * DD + col;
        float zval = out[o];              // Z (placed there by copy_kernel)
        out[o] = zval + kern[m] * (chg[col] + acc[t][r]);
      }
    }
  }
}

// ---------------------------------------------------------------------------
extern "C" void kernel_launch(void* const* d_in, const int* in_sizes, int n_in,
                              void* d_out, int out_size, void* d_ws, size_t ws_size,
                              hipStream_t stream) {
  const float* Zall = (const float*)d_in[0];
  const float* U    = (const float*)d_in[1];
  const float* V    = (const float*)d_in[2];
  const float* zmu  = (const float*)d_in[3];
  const float* xc   = (const float*)d_in[4];
  const int*   cidx = (const int*)d_in[5];
  float* out = (float*)d_out;
  float* wsf = (float*)d_ws;

  const int N = in_sizes[5];              // choice_index length == rows
  const long n4 = ((long)N * DD) / 4;

  copy_kernel<<<8192, 256, 0, stream>>>(Zall, out, n4);
  prep_kernel<<<1, 512, 0, stream>>>(U, zmu, xc, wsf);

  const int nblk = (N + 15) / 16;
  main_kernel<<<nblk, 256, 0, stream>>>(Zall, U, V, zmu, cidx, wsf, out, N);
}